// SignalPreprocess_56281251447193
// MI455X (gfx1250) — compile-verified
//
#include <hip/hip_runtime.h>

// ---------------------------------------------------------------------------
// SignalPreprocess: 4x fused [minpool3 -> rowwise minmax-norm -> pad -> avgpool3]
// Memory-bound (328 MB total HBM traffic, ~14us floor @ 23.3 TB/s).
// One workgroup per row; row lives in LDS for all 4 rounds (single HBM pass).
// Uses gfx1250 async global<->LDS DMA path (ASYNCcnt) for row staging.
// ---------------------------------------------------------------------------

#define EPSV 1e-9f

constexpr int W      = 5000;        // row width (floats)
constexpr int W4     = W / 4;       // 1250 float4 chunks (row is 16B aligned: 20000 % 16 == 0)
constexpr int NT     = 512;         // threads per block = 16 wave32
constexpr int NWAVES = NT / 32;     // 16
constexpr int PER    = (W  + NT - 1) / NT;   // 10 elements per thread
constexpr int PER4   = (W4 + NT - 1) / NT;   // 3 float4 per thread

// async-to-LDS builtins take int4 pointers: (v4i AS1*, v4i AS3*, imm off, imm cpol)
typedef int v4i __attribute__((ext_vector_type(4)));
typedef __attribute__((address_space(1))) v4i* gv4i_ptr;
typedef __attribute__((address_space(3))) v4i* lv4i_ptr;

#if defined(__has_builtin)
#if __has_builtin(__builtin_amdgcn_global_load_async_to_lds_b128)
#define HAVE_ASYNC_LD 1
#endif
#if __has_builtin(__builtin_amdgcn_global_store_async_from_lds_b128)
#define HAVE_ASYNC_ST 1
#endif
#if __has_builtin(__builtin_amdgcn_s_wait_asynccnt)
#define WAIT_ASYNC() __builtin_amdgcn_s_wait_asynccnt(0)
#endif
#endif
#ifndef WAIT_ASYNC
#define WAIT_ASYNC() asm volatile("s_wait_asynccnt 0" ::: "memory")
#endif

__global__ __launch_bounds__(NT)
void pool4_rows_kernel(const float* __restrict__ x, float* __restrict__ out) {
    __shared__ __align__(16) float A[W];        // ping
    __shared__ __align__(16) float B[W];        // pong (normalized minpool, tail zeroed)
    __shared__ float red[2 * NWAVES];           // [0..15]=per-wave min, [16..31]=per-wave max

    const int    t      = threadIdx.x;
    const size_t rowoff = (size_t)blockIdx.x * (size_t)W;
    const float* gin    = x + rowoff;
    float*       gout   = out + rowoff;

    // ---- stage row into LDS A (async DMA, no VGPR bounce) ----
#ifdef HAVE_ASYNC_LD
    #pragma unroll
    for (int k = 0; k < PER4; ++k) {
        int p4 = t + k * NT;
        if (p4 < W4) {
            __builtin_amdgcn_global_load_async_to_lds_b128(
                (gv4i_ptr)(gin + 4 * p4), (lv4i_ptr)(A + 4 * p4), 0, 0);
        }
    }
    WAIT_ASYNC();
#else
    #pragma unroll
    for (int k = 0; k < PER4; ++k) {
        int p4 = t + k * NT;
        if (p4 < W4) {
            float4 v = ((const float4*)gin)[p4];
            ((float4*)A)[p4] = v;
        }
    }
#endif
    __syncthreads();

    const int wave = t >> 5;
    const int lane = t & 31;

    #pragma unroll 1
    for (int blk = 0; blk < 4; ++blk) {
        // ---- minpool3 (width 4998) into registers + thread-local min/max ----
        float m[PER];
        float lmin =  3.402823466e+38f;
        float lmax = -3.402823466e+38f;
        #pragma unroll
        for (int k = 0; k < PER; ++k) {
            int p = t + k * NT;
            if (p < W - 2) {
                float v = fminf(fminf(A[p], A[p + 1]), A[p + 2]);
                m[k] = v;
                lmin = fminf(lmin, v);
                lmax = fmaxf(lmax, v);
            } else {
                m[k] = 0.0f;
            }
        }

        // ---- wave32 reduction, then cross-wave combine via LDS ----
        #pragma unroll
        for (int o = 16; o > 0; o >>= 1) {
            lmin = fminf(lmin, __shfl_xor(lmin, o, 32));
            lmax = fmaxf(lmax, __shfl_xor(lmax, o, 32));
        }
        if (lane == 0) {
            red[wave]          = lmin;
            red[NWAVES + wave] = lmax;
        }
        __syncthreads();
        float pmin = red[0];
        float pmax = red[NWAVES];
        #pragma unroll
        for (int w = 1; w < NWAVES; ++w) {
            pmin = fminf(pmin, red[w]);
            pmax = fmaxf(pmax, red[NWAVES + w]);
        }
        const float inv = 1.0f / (pmax - pmin);

        // ---- normalize into B; zero-pad last 2 ----
        #pragma unroll
        for (int k = 0; k < PER; ++k) {
            int p = t + k * NT;
            if (p < W - 2)      B[p] = (m[k] - pmin) * inv + EPSV;
            else if (p < W)     B[p] = 0.0f;
        }
        __syncthreads();

        // ---- avgpool3 (pad=1, count_include_pad) B -> A ----
        #pragma unroll
        for (int k = 0; k < PER; ++k) {
            int p = t + k * NT;
            if (p < W) {
                float lv = (p > 0)     ? B[p - 1] : 0.0f;
                float rv = (p < W - 1) ? B[p + 1] : 0.0f;
                A[p] = (lv + B[p] + rv) * (1.0f / 3.0f);
            }
        }
        __syncthreads();
    }

    // ---- stream result back (async LDS -> global) ----
#ifdef HAVE_ASYNC_ST
    #pragma unroll
    for (int k = 0; k < PER4; ++k) {
        int p4 = t + k * NT;
        if (p4 < W4) {
            __builtin_amdgcn_global_store_async_from_lds_b128(
                (gv4i_ptr)(gout + 4 * p4), (lv4i_ptr)(A + 4 * p4), 0, 0);
        }
    }
    WAIT_ASYNC();
#else
    #pragma unroll
    for (int k = 0; k < PER4; ++k) {
        int p4 = t + k * NT;
        if (p4 < W4) {
            ((float4*)gout)[p4] = ((const float4*)A)[p4];
        }
    }
#endif
}

extern "C" void kernel_launch(void* const* d_in, const int* in_sizes, int n_in,
                              void* d_out, int out_size, void* d_ws, size_t ws_size,
                              hipStream_t stream) {
    (void)n_in; (void)out_size; (void)d_ws; (void)ws_size;
    const float* x   = (const float*)d_in[0];
    float*       out = (float*)d_out;
    const int rows   = in_sizes[0] / W;   // 8192
    pool4_rows_kernel<<<rows, NT, 0, stream>>>(x, out);
}